// Scorer_14791867367764
// MI455X (gfx1250) — compile-verified
//
#include <hip/hip_runtime.h>
#include <hip/hip_bf16.h>

typedef __attribute__((ext_vector_type(2))) float v2f;
typedef __attribute__((ext_vector_type(8))) float v8f;

#define NPTS 128
#define FD   256
#define EPSV 1e-5f

// Bitonic sort of 256 (key,payload,idx) triples, descending by key,
// ties broken by ascending original index (matches stable argsort(-x)).
static __device__ inline void bitonic256_desc(float* skey, float* sval, int* sidx, int tid) {
    for (int k = 2; k <= 256; k <<= 1) {
        for (int j = k >> 1; j > 0; j >>= 1) {
            int p = tid ^ j;
            if (p > tid) {
                bool  up = ((tid & k) == 0);
                float ka = skey[tid], kb = skey[p];
                int   ia = sidx[tid], ib = sidx[p];
                bool  aBeforeB = (ka > kb) || (ka == kb && ia < ib);
                if (up != aBeforeB) {
                    skey[tid] = kb; skey[p] = ka;
                    sidx[tid] = ib; sidx[p] = ia;
                    float va = sval[tid]; sval[tid] = sval[p]; sval[p] = va;
                }
            }
            __syncthreads();
        }
    }
}

__global__ __launch_bounds__(256) void scorer_fused_kernel(
    const float* __restrict__ points_c,
    const float* __restrict__ ref_feats,
    const float* __restrict__ src_feats,
    const float* __restrict__ trans,
    const float* __restrict__ W1, const float* __restrict__ b1,
    const float* __restrict__ g1, const float* __restrict__ bt1,
    const float* __restrict__ W2, const float* __restrict__ b2,
    const float* __restrict__ g2, const float* __restrict__ bt2,
    const float* __restrict__ W3, const float* __restrict__ b3,
    float* __restrict__ out,
    float* __restrict__ dist_mat,   // ws: 128*128 floats
    float* __restrict__ feat_mat)   // ws: 128*128 floats
{
    const int tid = threadIdx.x;    // 0..255

    __shared__ float pts[2 * NPTS][3];
    __shared__ float centroid[3];
    __shared__ float maxn;
    __shared__ float md_ref[NPTS], md_src[NPTS];
    __shared__ int   mi_ref[NPTS], mi_src[NPTS];
    __shared__ float fm_ref[NPTS], fm_src[NPTS];
    __shared__ int   fi_ref[NPTS], fi_src[NPTS];
    __shared__ float skey[256], sval[256];
    __shared__ int   sidx[256];
    __shared__ float geo[512];
    __shared__ float h1[256];
    __shared__ float h2[128];

    // ---------- Stage A: transform src points, normalize point cloud ----------
    {
        float x = points_c[tid * 3 + 0];
        float y = points_c[tid * 3 + 1];
        float z = points_c[tid * 3 + 2];
        if (tid >= NPTS) {  // src: p' = R p + t  (R = trans[:3,:3], row-major 4x4)
            float nx = trans[0] * x + trans[1] * y + trans[2]  * z + trans[3];
            float ny = trans[4] * x + trans[5] * y + trans[6]  * z + trans[7];
            float nz = trans[8] * x + trans[9] * y + trans[10] * z + trans[11];
            x = nx; y = ny; z = nz;
        }
        pts[tid][0] = x; pts[tid][1] = y; pts[tid][2] = z;
    }
    __syncthreads();
    if (tid < 3) {
        float s = 0.f;
        for (int i = 0; i < 2 * NPTS; ++i) s += pts[i][tid];
        centroid[tid] = s * (1.f / (2 * NPTS));
    }
    __syncthreads();
    pts[tid][0] -= centroid[0];
    pts[tid][1] -= centroid[1];
    pts[tid][2] -= centroid[2];
    __syncthreads();
    if (tid == 0) {
        float m = 0.f;
        for (int i = 0; i < 2 * NPTS; ++i) {
            float nn = sqrtf(pts[i][0] * pts[i][0] + pts[i][1] * pts[i][1] +
                             pts[i][2] * pts[i][2]);
            m = fmaxf(m, nn);
        }
        maxn = m;
    }
    __syncthreads();
    {
        float inv = 1.f / maxn;
        pts[tid][0] *= inv; pts[tid][1] *= inv; pts[tid][2] *= inv;
    }
    __syncthreads();

    // ---------- Stage B: dist_mat[i][j] = exp(-||ref_i - src_j||^2) ----------
    for (int e = tid; e < NPTS * NPTS; e += 256) {
        int i = e >> 7, j = e & 127;
        float dx = pts[i][0] - pts[NPTS + j][0];
        float dy = pts[i][1] - pts[NPTS + j][1];
        float dz = pts[i][2] - pts[NPTS + j][2];
        float d  = dx * dx + dy * dy + dz * dz;
        dist_mat[e] = expf(-d);
    }

    // ---------- Stage C: feat_mat = ref_feats @ src_feats^T via V_WMMA_F32_16X16X4_F32 ----------
    {
        const int wave = tid >> 5;          // 0..7
        const int lane = tid & 31;
        const int lm   = lane & 15;         // M (for A) / N (for B) within tile
        const int kh   = lane >> 4;         // lane-half selects K pair
        for (int t = wave; t < 64; t += 8) {
            const int m0 = (t >> 3) << 4;
            const int n0 = (t & 7) << 4;
            v8f acc = {};
            const float* __restrict__ arow = ref_feats + (m0 + lm) * FD;
            const float* __restrict__ brow = src_feats + (n0 + lm) * FD;  // B[k][n] = src[n][k]
            for (int kb = 0; kb < FD; kb += 4) {
                v2f a, b;
                a.x = arow[kb + 2 * kh + 0];
                a.y = arow[kb + 2 * kh + 1];
                b.x = brow[kb + 2 * kh + 0];
                b.y = brow[kb + 2 * kh + 1];
                acc = __builtin_amdgcn_wmma_f32_16x16x4_f32(
                    /*neg_a=*/false, a, /*neg_b=*/false, b,
                    /*c_mod=*/(short)0, acc, /*reuse_a=*/false, /*reuse_b=*/false);
            }
            // D layout: VGPR r -> rows m0+r (lanes 0..15) / m0+8+r (lanes 16..31), col n0+lm
            const int rbase = m0 + (kh << 3);
#pragma unroll
            for (int r = 0; r < 8; ++r)
                feat_mat[(rbase + r) * NPTS + (n0 + lm)] = acc[r];
        }
    }
    __threadfence();   // make ws writes visible to all waves in the block
    __syncthreads();

    // ---------- Stage D: row/col max + argmax of both matrices ----------
    if (tid < NPTS) {
        int i = tid;
        float bd = -1.f; int bdi = 0; float bf = -1e30f; int bfi = 0;
        for (int j = 0; j < NPTS; ++j) {
            float d = dist_mat[i * NPTS + j];
            if (d > bd) { bd = d; bdi = j; }
            float f = feat_mat[i * NPTS + j];
            if (f > bf) { bf = f; bfi = j; }
        }
        md_ref[i] = bd; mi_ref[i] = bdi; fm_ref[i] = bf; fi_ref[i] = bfi;
    } else {
        int j = tid - NPTS;
        float bd = -1.f; int bdi = 0; float bf = -1e30f; int bfi = 0;
        for (int i = 0; i < NPTS; ++i) {
            float d = dist_mat[i * NPTS + j];
            if (d > bd) { bd = d; bdi = i; }
            float f = feat_mat[i * NPTS + j];
            if (f > bf) { bf = f; bfi = i; }
        }
        md_src[j] = bd; mi_src[j] = bdi; fm_src[j] = bf; fi_src[j] = bfi;
    }
    __syncthreads();

    // ---------- Stage E/F: sort #1 (min_dist desc, payload = feat_score) ----------
    {
        float key, val;
        if (tid < NPTS) { key = md_ref[tid]; val = feat_mat[tid * NPTS + mi_ref[tid]]; }
        else { int j = tid - NPTS; key = md_src[j]; val = feat_mat[mi_src[j] * NPTS + j]; }
        skey[tid] = key; sval[tid] = val; sidx[tid] = tid;
    }
    __syncthreads();
    bitonic256_desc(skey, sval, sidx, tid);
    geo[tid] = skey[tid] * sval[tid];           // d_m
    __syncthreads();

    // sort #2 (match_feat desc, payload = match_dist)
    {
        float key, val;
        if (tid < NPTS) { key = fm_ref[tid]; val = dist_mat[tid * NPTS + fi_ref[tid]]; }
        else { int j = tid - NPTS; key = fm_src[j]; val = dist_mat[fi_src[j] * NPTS + j]; }
        skey[tid] = key; sval[tid] = val; sidx[tid] = tid;
    }
    __syncthreads();
    bitonic256_desc(skey, sval, sidx, tid);
    geo[256 + tid] = skey[tid] * sval[tid];     // m_d
    __syncthreads();

    // ---------- Stage G: MLP ----------
    // layer 1: (1,512) @ W1^T -> (1,256)
    {
        float s = b1[tid];
        const float* __restrict__ w = W1 + tid * 512;
        for (int k = 0; k < 512; ++k) s += geo[k] * w[k];
        h1[tid] = s;
    }
    __syncthreads();
    // group norm (8 groups of 32) + relu
    {
        int base = (tid >> 5) << 5;
        float m = 0.f;
        for (int k = 0; k < 32; ++k) m += h1[base + k];
        m *= (1.f / 32.f);
        float v = 0.f;
        for (int k = 0; k < 32; ++k) { float dlt = h1[base + k] - m; v += dlt * dlt; }
        v *= (1.f / 32.f);
        float xn = (h1[tid] - m) / sqrtf(v + EPSV);
        float hv = fmaxf(xn * g1[tid] + bt1[tid], 0.f);
        __syncthreads();
        h1[tid] = hv;
    }
    __syncthreads();
    // layer 2: (1,256) @ W2^T -> (1,128)
    if (tid < 128) {
        float s = b2[tid];
        const float* __restrict__ w = W2 + tid * 256;
        for (int k = 0; k < 256; ++k) s += h1[k] * w[k];
        h2[tid] = s;
    }
    __syncthreads();
    // group norm (8 groups of 16) + relu
    float hv2 = 0.f;
    if (tid < 128) {
        int base = (tid >> 4) << 4;
        float m = 0.f;
        for (int k = 0; k < 16; ++k) m += h2[base + k];
        m *= (1.f / 16.f);
        float v = 0.f;
        for (int k = 0; k < 16; ++k) { float dlt = h2[base + k] - m; v += dlt * dlt; }
        v *= (1.f / 16.f);
        float xn = (h2[tid] - m) / sqrtf(v + EPSV);
        hv2 = fmaxf(xn * g2[tid] + bt2[tid], 0.f);
    }
    __syncthreads();
    if (tid < 128) h2[tid] = hv2;
    __syncthreads();
    // layer 3: (1,128) @ W3^T -> (1,2)
    if (tid < 2) {
        float s = b3[tid];
        const float* __restrict__ w = W3 + tid * 128;
        for (int k = 0; k < 128; ++k) s += h2[k] * w[k];
        out[tid] = s;
    }
}

extern "C" void kernel_launch(void* const* d_in, const int* in_sizes, int n_in,
                              void* d_out, int out_size, void* d_ws, size_t ws_size,
                              hipStream_t stream) {
    (void)in_sizes; (void)n_in; (void)out_size; (void)ws_size;
    const float* points = (const float*)d_in[0];
    const float* reff   = (const float*)d_in[1];
    const float* srcf   = (const float*)d_in[2];
    const float* trans  = (const float*)d_in[3];
    const float* W1  = (const float*)d_in[4];
    const float* b1  = (const float*)d_in[5];
    const float* g1  = (const float*)d_in[6];
    const float* bt1 = (const float*)d_in[7];
    const float* W2  = (const float*)d_in[8];
    const float* b2  = (const float*)d_in[9];
    const float* g2  = (const float*)d_in[10];
    const float* bt2 = (const float*)d_in[11];
    const float* W3  = (const float*)d_in[12];
    const float* b3  = (const float*)d_in[13];

    float* dist_mat = (float*)d_ws;               // 128*128 floats
    float* feat_mat = dist_mat + NPTS * NPTS;     // 128*128 floats

    scorer_fused_kernel<<<1, 256, 0, stream>>>(
        points, reff, srcf, trans,
        W1, b1, g1, bt1, W2, b2, g2, bt2, W3, b3,
        (float*)d_out, dist_mat, feat_mat);
}